// MCMCSampler_80822694576460
// MI455X (gfx1250) — compile-verified
//
#include <hip/hip_runtime.h>

// ---------------------------------------------------------------------------
// MCMC sampler (MI455X / gfx1250, wave32).
//
// feat_energy = einsum(features, W_feat) and the bias have ZERO gradient
// w.r.t. predictions, and the output is only the per-step predictions. So
// features/W_feat/b are dead inputs: skip 48 MB of reads and the matvec.
// The live computation is an elementwise 16-step recurrence; it is
// transcendental/VALU bound (store floor 100MB/23.3TB/s ~= 4.3us), so the
// optimization target is minimum trans ops per step:
//   - sigmoid via v_tanh_f32 (1 trans instead of exp+rcp)
//   - s/(s+e) - (1-s)/(1-s+e) == e*tanh(p/2) / ((s+e)(1-s+e))  -> 1 rcp
//   => 4 trans (tanh, 2x log2, rcp) + ~13 VALU per element-step.
//
// NOTE: mcmc_sampler_kernel is defined FIRST so the compile-loop's asm
// snippet shows its codegen (v_tanh_f32, NT b128 loads/stores).
// ---------------------------------------------------------------------------

typedef float v4f __attribute__((ext_vector_type(4)));

namespace {
constexpr int   kV     = 64;     // NUM_VARIANTS
constexpr int   kB     = 1024;
constexpr int   kH     = 24;
constexpr int   kSteps = 16;     // MCMC_STEPS
constexpr int   kBH    = kB * kH;              // 24576
constexpr int   kN     = kV * kB * kH;         // 1,572,864 elements per step
constexpr float kStep  = 0.1f;   // STEP_SIZE
constexpr float kBeta  = 0.1f;   // ENTROPY_BETA
constexpr float kEps   = 1e-7f;  // EPS
constexpr float kLog2E = 1.4426950408889634f;
constexpr float kLn2   = 0.6931471805599453f;
}

// One recurrence step for one element. 4 transcendental ops on the tanh
// path: v_tanh_f32, 2x v_log_f32, v_rcp_f32.
__device__ __forceinline__ float mcmc_step_elem(float p, float c, float wp) {
#if __has_builtin(__builtin_amdgcn_tanhf)
    const float th  = __builtin_amdgcn_tanhf(0.5f * p);      // tanh(p/2) = 2s-1
    const float s   = __builtin_fmaf( 0.5f, th, 0.5f);       // sigmoid(p)
    const float oms = __builtin_fmaf(-0.5f, th, 0.5f);       // 1 - s
#else
    const float t   = __builtin_amdgcn_exp2f(-p * kLog2E);   // exp(-p)
    const float s   = __builtin_amdgcn_rcpf(1.0f + t);       // sigmoid(p)
    const float oms = t * s;                                 // 1 - s
    const float th  = s - oms;                               // 2s - 1
#endif
    const float sp  = s * oms;                               // sigma'(p)
    const float a   = s   + kEps;
    const float b2  = oms + kEps;
    const float la  = __builtin_amdgcn_logf(a);              // log2(s+eps)
    const float lb  = __builtin_amdgcn_logf(b2);             // log2(1-s+eps)
    // s/(s+e) - (1-s)/(1-s+e) = e*(2s-1) / ((s+e)*(1-s+e))   (exact)
    const float rp    = __builtin_amdgcn_rcpf(a * b2);
    const float ratio = (kEps * th) * rp;
    // -dH/ds = ln((s+e)/(1-s+e)) + ratio
    const float negdH = __builtin_fmaf(la - lb, kLn2, ratio);
    // d(total)/dp (without shared coef) = sp * (wp - beta*dHds)
    const float g = sp * __builtin_fmaf(kBeta, negdH, wp);
    return __builtin_fmaf(-c, g, p);   // c = STEP * mask / (hpr * V * B)
}

// Main kernel: 4 contiguous elements/thread (H=24 % 4 == 0 -> a float4 never
// crosses a (v,b) row). 16 steps fully unrolled in registers; one
// non-temporal b128 store per step (output is write-once -> keep the 100 MB
// stream out of the 192 MB L2).
__global__ void __launch_bounds__(256)
mcmc_sampler_kernel(const float* __restrict__ pred0,
                    const float* __restrict__ coef,
                    const float* __restrict__ w_prob_ptr,
                    float* __restrict__ out) {
    const int tid = blockIdx.x * blockDim.x + threadIdx.x;
    const int idx = tid * 4;                 // element index in [0, kN)
    if (idx >= kN) return;

    const float wp = w_prob_ptr[0];          // uniform -> scalar load

    v4f p = __builtin_nontemporal_load((const v4f*)(pred0 + idx));   // read-once
    const int cidx = idx % kBH;              // same (b,h) for all 64 variants
    const v4f c = *(const v4f*)(coef + cidx);                        // L2-hot

#pragma unroll
    for (int s = 0; s < kSteps; ++s) {
#pragma unroll
        for (int e = 0; e < 4; ++e) p[e] = mcmc_step_elem(p[e], c[e], wp);
        // output layout [S,V,B,H,1] -> flat offset s*kN + idx
        __builtin_nontemporal_store(p, (v4f*)(out + (size_t)s * (size_t)kN + idx));
    }
}

// Prepass: coef[b,h] = STEP * mask[b,h] / (hpr[b] * V * B)
// (mask*mask == mask since mask is 0/1, so the gradient's second mask
//  multiply folds in for free). 98 KB result in d_ws, L2-hot, reused by all
//  64 variants.
__global__ void mcmc_coef_kernel(const int* __restrict__ amask,
                                 float* __restrict__ coef) {
    const int i = blockIdx.x * blockDim.x + threadIdx.x;
    if (i >= kBH) return;
    const int b = i / kH;
    const int* row = amask + b * kH;
    int hsum = 0;
#pragma unroll
    for (int j = 0; j < kH; ++j) hsum += row[j];
    const float m = (float)amask[i];
    coef[i] = kStep * m / ((float)hsum * (float)kV * (float)kB);
}

extern "C" void kernel_launch(void* const* d_in, const int* in_sizes, int n_in,
                              void* d_out, int out_size, void* d_ws, size_t ws_size,
                              hipStream_t stream) {
    (void)in_sizes; (void)n_in; (void)out_size; (void)ws_size;
    // setup_inputs order: features[0](dead), predictions_init[1], W_feat[2](dead),
    //                     w_prob[3], b[4](dead), attention_mask[5]
    const float* pred0  = (const float*)d_in[1];
    const float* w_prob = (const float*)d_in[3];
    const int*   amask  = (const int*)d_in[5];
    float*       coef   = (float*)d_ws;          // kBH floats = 98 KB scratch
    float*       out    = (float*)d_out;

    mcmc_coef_kernel<<<(kBH + 255) / 256, 256, 0, stream>>>(amask, coef);

    const int nThreads = kN / 4;                 // 393,216
    mcmc_sampler_kernel<<<nThreads / 256, 256, 0, stream>>>(pred0, coef, w_prob, out);
}